// DualPathRankingLoss_54348516163734
// MI455X (gfx1250) — compile-verified
//
#include <hip/hip_runtime.h>
#include <stdint.h>

#define B_ROWS   1024
#define N_COLS   5000
#define N_PAD    5120
#define NTHREADS 256
#define NWAVES   (NTHREADS / 32)
#define F4PT     (N_PAD / 4 / NTHREADS)   // 5 float4 chunks per thread
#define VPT      (F4PT * 4)               // 20 elements per thread
#define TOPK     10
#define SENT     1.0e9f
#define FLT_BIG  3.4e38f

typedef __attribute__((ext_vector_type(8))) float v8f;
typedef __attribute__((ext_vector_type(2))) float v2f;

__device__ __forceinline__ float softplus_f(float x) {
    // stable: max(x,0) + log1p(exp(-|x|))
    return fmaxf(x, 0.0f) + log1pf(expf(-fabsf(x)));
}

__device__ __forceinline__ float wave_sum32(float v) {
#pragma unroll
    for (int o = 16; o; o >>= 1) v += __shfl_xor(v, o, 32);
    return v;
}
__device__ __forceinline__ float wave_max32(float v) {
#pragma unroll
    for (int o = 16; o; o >>= 1) v = fmaxf(v, __shfl_xor(v, o, 32));
    return v;
}

__device__ float block_sum(float v, float* sc) {
    v = wave_sum32(v);
    int w = threadIdx.x >> 5, l = threadIdx.x & 31;
    if (l == 0) sc[w] = v;
    __syncthreads();
    if (threadIdx.x < 32) {
        float x = (threadIdx.x < NWAVES) ? sc[threadIdx.x] : 0.0f;
        x = wave_sum32(x);
        if (threadIdx.x == 0) sc[0] = x;
    }
    __syncthreads();
    float r = sc[0];
    __syncthreads();
    return r;
}

__device__ float block_max(float v, float* sc) {
    v = wave_max32(v);
    int w = threadIdx.x >> 5, l = threadIdx.x & 31;
    if (l == 0) sc[w] = v;
    __syncthreads();
    if (threadIdx.x < 32) {
        float x = (threadIdx.x < NWAVES) ? sc[threadIdx.x] : -FLT_BIG;
        x = wave_max32(x);
        if (threadIdx.x == 0) sc[0] = x;
    }
    __syncthreads();
    float r = sc[0];
    __syncthreads();
    return r;
}

// Block-wide arg-extremum with stable-sort tie rule (smaller index wins).
__device__ void block_best(float& v, int& i, bool maximize, float* scv, int* sci) {
#pragma unroll
    for (int o = 16; o; o >>= 1) {
        float ov = __shfl_xor(v, o, 32);
        int   oi = __shfl_xor(i, o, 32);
        bool take = maximize ? (ov > v || (ov == v && oi < i))
                             : (ov < v || (ov == v && oi < i));
        if (take) { v = ov; i = oi; }
    }
    int w = threadIdx.x >> 5, l = threadIdx.x & 31;
    if (l == 0) { scv[w] = v; sci[w] = i; }
    __syncthreads();
    if (threadIdx.x < 32) {
        float x; int xi;
        if (threadIdx.x < NWAVES) { x = scv[threadIdx.x]; xi = sci[threadIdx.x]; }
        else { x = maximize ? -FLT_BIG : FLT_BIG; xi = 0x7fffffff; }
#pragma unroll
        for (int o = 16; o; o >>= 1) {
            float ox = __shfl_xor(x, o, 32);
            int   oxi = __shfl_xor(xi, o, 32);
            bool take = maximize ? (ox > x || (ox == x && oxi < xi))
                                 : (ox < x || (ox == x && oxi < xi));
            if (take) { x = ox; xi = oxi; }
        }
        if (threadIdx.x == 0) { scv[0] = x; sci[0] = xi; }
    }
    __syncthreads();
    v = scv[0]; i = sci[0];
    __syncthreads();
}

__global__ __launch_bounds__(NTHREADS)
void row_loss_kernel(const float* __restrict__ up, const float* __restrict__ dn,
                     const float* __restrict__ yt, const int* __restrict__ mk,
                     float* __restrict__ rowloss) {
    __shared__ __align__(16) float y_s[N_PAD];   // masked y (invalid -> -SENT)
    __shared__ __align__(16) float up_s[N_PAD];
    __shared__ __align__(16) float dn_s[N_PAD];  // holds mask first, then dn
    __shared__ float scv[32];
    __shared__ int   sci[32];

    const int tid = threadIdx.x;
    const int row = blockIdx.x;
    const size_t base = (size_t)row * N_COLS;

    uint32_t lds_y = (uint32_t)(uintptr_t)&y_s[0];
    uint32_t lds_u = (uint32_t)(uintptr_t)&up_s[0];
    uint32_t lds_d = (uint32_t)(uintptr_t)&dn_s[0];
    uint64_t g_y = (uint64_t)(uintptr_t)(yt + base);
    uint64_t g_u = (uint64_t)(uintptr_t)(up + base);
    uint64_t g_d = (uint64_t)(uintptr_t)(dn + base);
    uint64_t g_m = (uint64_t)(uintptr_t)(mk + base);

    const int NCHUNK = (N_COLS * 4) / 16;  // 1250 x 16B (row stride 16B aligned)
    // Stage y, up, mask(->dn_s temp) via CDNA5 async load-to-LDS (ASYNCcnt path)
    for (int t = tid; t < NCHUNK; t += NTHREADS) {
        uint32_t off = (uint32_t)t * 16u;
        asm volatile("global_load_async_to_lds_b128 %0, %1, off"
                     :: "v"(lds_y + off), "v"(g_y + off) : "memory");
        asm volatile("global_load_async_to_lds_b128 %0, %1, off"
                     :: "v"(lds_u + off), "v"(g_u + off) : "memory");
        asm volatile("global_load_async_to_lds_b128 %0, %1, off"
                     :: "v"(lds_d + off), "v"(g_m + off) : "memory");
    }
    // Zero the mask pad region (async writes only bytes [0,20000)).
    if (tid < N_PAD - N_COLS) ((int*)dn_s)[N_COLS + tid] = 0;

    asm volatile("s_wait_asynccnt 0" ::: "memory");
    __syncthreads();

    // Pass A (fused with mask-fold): each thread owns elements 4*(tid+256*j)+c.
    // Cache masked y in registers; write sentinel-masked y back to LDS.
    float yreg[VPT];
    float cnt = 0.0f, my = -FLT_BIG, mu = -FLT_BIG, ssp_u = 0.0f;
#pragma unroll
    for (int j = 0; j < F4PT; ++j) {
        int f = tid + NTHREADS * j;
        int4   m4 = ((const int4*)dn_s)[f];
        float4 y4 = ((const float4*)y_s)[f];
        float4 u4 = ((const float4*)up_s)[f];
        int   ma[4] = {m4.x, m4.y, m4.z, m4.w};
        float ya[4] = {y4.x, y4.y, y4.z, y4.w};
        float ua[4] = {u4.x, u4.y, u4.z, u4.w};
#pragma unroll
        for (int c = 0; c < 4; ++c) {
            float ym = (ma[c] > 0) ? ya[c] : -SENT;
            yreg[4 * j + c] = ym;
            if (ym != -SENT) {
                cnt += 1.0f;
                my = fmaxf(my, ym);
                mu = fmaxf(mu, ua[c]);
                ssp_u += softplus_f(ua[c]);
            }
            ya[c] = ym;
        }
        float4 w4; w4.x = ya[0]; w4.y = ya[1]; w4.z = ya[2]; w4.w = ya[3];
        ((float4*)y_s)[f] = w4;
    }
    __syncthreads();  // all mask reads done; y_s fully sentinel-masked

    // Overlap: start dn DMA, run pass-A reductions meanwhile.
    for (int t = tid; t < NCHUNK; t += NTHREADS) {
        uint32_t off = (uint32_t)t * 16u;
        asm volatile("global_load_async_to_lds_b128 %0, %1, off"
                     :: "v"(lds_d + off), "v"(g_d + off) : "memory");
    }
    float nv = block_sum(cnt, scv);
    my = block_max(my, scv);
    mu = block_max(mu, scv);
    ssp_u = block_sum(ssp_u, scv);

    asm volatile("s_wait_asynccnt 0" ::: "memory");
    __syncthreads();  // dn_s now holds down_logits

    // Pass B: softplus(dn) sum, softmax partitions, q*(y-up) numerator.
    float ssp_d = 0.0f, zy = 0.0f, zu = 0.0f, tq = 0.0f;
#pragma unroll
    for (int j = 0; j < F4PT; ++j) {
        int f = tid + NTHREADS * j;
        float4 u4 = ((const float4*)up_s)[f];
        float4 d4 = ((const float4*)dn_s)[f];
        float ua[4] = {u4.x, u4.y, u4.z, u4.w};
        float da[4] = {d4.x, d4.y, d4.z, d4.w};
#pragma unroll
        for (int c = 0; c < 4; ++c) {
            float ym = yreg[4 * j + c];
            if (ym != -SENT) {
                ssp_d += softplus_f(da[c]);
                float ey = expf(ym - my);
                zy += ey;
                zu += expf(ua[c] - mu);
                tq += ey * (ym - ua[c]);
            }
        }
    }
    ssp_d = block_sum(ssp_d, scv);
    zy = block_sum(zy, scv);
    zu = block_sum(zu, scv);
    tq = block_sum(tq, scv);

    int k = (int)fminf((float)TOPK, nv);

    // Top-k by y (desc, stable ties by index) scanned from registers.
    float sum_up_top = 0.0f, sum_dn_bot = 0.0f;
    float pv = FLT_BIG; int pi = -1;
    for (int t = 0; t < k; ++t) {
        float bv = -FLT_BIG; int bi = 0x7fffffff;
#pragma unroll
        for (int q = 0; q < VPT; ++q) {
            int j = q >> 2, c = q & 3;
            int e = 4 * (tid + NTHREADS * j) + c;
            float v = yreg[q];  // pads are -SENT: behave like masked-out entries
            if ((v < pv || (v == pv && e > pi)) &&
                (v > bv || (v == bv && e < bi))) { bv = v; bi = e; }
        }
        block_best(bv, bi, true, scv, sci);
        sum_up_top += up_s[bi];
        pv = bv; pi = bi;
    }
    // Bottom-k by y (asc, stable ties by index).
    pv = -FLT_BIG; pi = -1;
    for (int t = 0; t < k; ++t) {
        float bv = FLT_BIG; int bi = 0x7fffffff;
#pragma unroll
        for (int q = 0; q < VPT; ++q) {
            int j = q >> 2, c = q & 3;
            int e = 4 * (tid + NTHREADS * j) + c;
            float y = yreg[q];
            float v = (y == -SENT) ? SENT : y;   // reference +NEG fill
            if ((v > pv || (v == pv && e > pi)) &&
                (v < bv || (v == bv && e < bi))) { bv = v; bi = e; }
        }
        block_best(bv, bi, false, scv, sci);
        sum_dn_bot += dn_s[bi];
        pv = bv; pi = bi;
    }

    if (tid == 0) {
        float L = 0.0f;
        if (nv > 0.0f) {
            float up_loss = (ssp_u - sum_up_top) / nv;
            float dn_loss = (ssp_d - sum_dn_bot) / nv;
            float lse_y = my + logf(zy);
            float lse_u = mu + logf(zu);
            float kl = (tq / zy + (lse_u - lse_y)) / nv;
            L = 1.0f * up_loss + 0.5f * dn_loss + 0.3f * kl;
        }
        rowloss[row] = L;
    }
}

// Deterministic 1024 -> 1 reduction using the matrix pipe:
// A(16x4) holds the 32 lane partials (K=0 slot for lanes 0-15, K=2 for 16-31),
// B = ones(4x16)  =>  D[m][n] = p[m] + p[m+16]; sum D rows, fold halves.
__global__ __launch_bounds__(32)
void final_reduce_kernel(const float* __restrict__ rowloss, float* __restrict__ out) {
    int lane = threadIdx.x;
    float s = 0.0f;
#pragma unroll
    for (int j = 0; j < B_ROWS / 32; ++j) s += rowloss[lane + 32 * j];

    v2f a; a[0] = s;    a[1] = 0.0f;
    v2f b; b[0] = 1.0f; b[1] = 1.0f;
    v8f c = {};
    v8f d = __builtin_amdgcn_wmma_f32_16x16x4_f32(false, a, false, b,
                                                  (short)0, c, false, false);
    float t = d[0] + d[1] + d[2] + d[3] + d[4] + d[5] + d[6] + d[7];
    t += __shfl_xor(t, 16, 32);   // fold the other half's row-sum block
    if (lane == 0) out[0] = t * (1.0f / (float)B_ROWS);
}

extern "C" void kernel_launch(void* const* d_in, const int* in_sizes, int n_in,
                              void* d_out, int out_size, void* d_ws, size_t ws_size,
                              hipStream_t stream) {
    const float* up = (const float*)d_in[0];
    const float* dn = (const float*)d_in[1];
    const float* yt = (const float*)d_in[2];
    const int*   mk = (const int*)d_in[3];
    float* rowloss = (float*)d_ws;

    row_loss_kernel<<<B_ROWS, NTHREADS, 0, stream>>>(up, dn, yt, mk, rowloss);
    final_reduce_kernel<<<1, 32, 0, stream>>>(rowloss, (float*)d_out);
}